// ColorQuantization_40673340293273
// MI455X (gfx1250) — compile-verified
//
#include <hip/hip_runtime.h>
#include <hip/hip_bf16.h>
#include <stdint.h>

// ---- problem constants (from reference: x is (32, 3, 512, 512) f32) ----
#define HW      262144              // 512*512
#define NBATCH  32
#define CHUNK   1024                // pixels per TDM tile per channel
#define NCHUNKS (NBATCH * (HW / CHUNK))   // 8192
#define THREADS 256                 // 8 wave32 waves
#define GRID    2048                // 4 chunks per block -> double buffering pays off

typedef __attribute__((ext_vector_type(4))) float        f32x4;
typedef __attribute__((ext_vector_type(4))) unsigned int u32x4;
typedef __attribute__((ext_vector_type(8))) int          i32x8;
typedef __attribute__((ext_vector_type(4))) int          i32x4;

// LDS byte offset of a __shared__ object (generic -> AS(3) -> int)
__device__ __forceinline__ uint32_t lds_offset(const void* p) {
    return (uint32_t)(uintptr_t)(__attribute__((address_space(3))) const void*)p;
}

// Issue one TDM load: (3 x CHUNK) f32 tile, row stride = HW elements, into LDS.
// D# layout per CDNA5 ISA 8.3/8.4. Groups 2/3 zero (2D tensor, tile_dim2..4 = 0).
__device__ __forceinline__ void issue_tdm_load(const float* src_tile, uint32_t lds_byte_addr) {
    uint64_t ga = (uint64_t)(uintptr_t)src_tile;
    u32x4 g0;
    g0[0] = 1u;                                        // count=1, user descriptor
    g0[1] = lds_byte_addr;                             // lds_addr [63:32]
    g0[2] = (uint32_t)(ga & 0xFFFFFFFFu);              // global_addr [95:64]
    g0[3] = (uint32_t)((ga >> 32) & 0x01FFFFFFu)       // global_addr [120:96]
          | (2u << 30);                                // type=2 ("image") [127:126]
    i32x8 g1;
    g1[0] = (int)(2u << 16);                           // workgroup_mask=0, data_size=2 (4B)
    g1[1] = (int)((unsigned)CHUNK << 16);              // tensor_dim0[15:0] @ bits 63:48
    g1[2] = (int)(3u << 16);                           // tensor_dim0[31:16]=0, tensor_dim1=3
    g1[3] = (int)((unsigned)CHUNK << 16);              // tensor_dim1[31:16]=0, tile_dim0=CHUNK
    g1[4] = 3;                                         // tile_dim1=3, tile_dim2=0
    g1[5] = (int)HW;                                   // tensor_dim0_stride[31:0] = HW
    g1[6] = 0;                                         // stride[47:32]=0, dim1_stride lo=0
    g1[7] = 0;                                         // dim1_stride hi (unused for 2D)
    i32x4 gz4 = {0, 0, 0, 0};
    i32x8 gz8 = {0, 0, 0, 0, 0, 0, 0, 0};
    __builtin_amdgcn_tensor_load_to_lds(g0, g1, gz4, gz4, gz8, 0);
}

// Soft color quantization for one pixel.
// weights ∝ exp(20 * x·c_j)  (‖x‖² and ‖c_j‖²=3 cancel in softmax)
// out_k = 2*w_{k+1} - 1
__device__ __forceinline__ void quant3(float x0, float x1, float x2,
                                       float& o0, float& o1, float& o2) {
    float s  = x0 + x1 + x2;
    float l0 = -s;
    float l1 = 2.0f * x0 - s;
    float l2 = 2.0f * x1 - s;
    float l3 = 2.0f * x2 - s;
    float m  = fmaxf(fmaxf(l0, l1), fmaxf(l2, l3));
    const float C = 28.853900817779268f;               // (2/T)*log2(e), T=0.1
    float e0 = __builtin_amdgcn_exp2f(C * (l0 - m));
    float e1 = __builtin_amdgcn_exp2f(C * (l1 - m));
    float e2 = __builtin_amdgcn_exp2f(C * (l2 - m));
    float e3 = __builtin_amdgcn_exp2f(C * (l3 - m));
    float r  = __builtin_amdgcn_rcpf(e0 + e1 + e2 + e3);
    float r2 = r + r;
    o0 = __builtin_fmaf(e1, r2, -1.0f);
    o1 = __builtin_fmaf(e2, r2, -1.0f);
    o2 = __builtin_fmaf(e3, r2, -1.0f);
}

__global__ __launch_bounds__(THREADS)
void ColorQuantization_kernel(const float* __restrict__ in, float* __restrict__ out) {
    __shared__ float buf[2][3 * CHUNK];                // 24 KB double buffer
    const int  tid    = (int)threadIdx.x;
    const bool issuer = (tid < 32);                    // wave 0 drives the TDM

    int t = (int)blockIdx.x;
    if (issuer && t < NCHUNKS) {
        int b = t >> 8;                                // 256 chunks per image
        int col = (t & 255) * CHUNK;
        issue_tdm_load(in + (size_t)b * 3 * HW + col, lds_offset(&buf[0][0]));
    }

    int cur = 0;
    for (; t < NCHUNKS; t += GRID) {
        int tn = t + GRID;
        if (issuer) {
            if (tn < NCHUNKS) {
                int bn = tn >> 8;
                int cn = (tn & 255) * CHUNK;
                issue_tdm_load(in + (size_t)bn * 3 * HW + cn, lds_offset(&buf[cur ^ 1][0]));
                __builtin_amdgcn_s_wait_tensorcnt(1);  // oldest DMA (cur buffer) done
            } else {
                __builtin_amdgcn_s_wait_tensorcnt(0);  // drain
            }
        }
        __syncthreads();                               // publish DMA'd tile to all waves

        const f32x4* lp = (const f32x4*)&buf[cur][0];
        f32x4 a0 = lp[tid];                            // channel 0, 4 pixels
        f32x4 a1 = lp[(CHUNK / 4) + tid];              // channel 1
        f32x4 a2 = lp[2 * (CHUNK / 4) + tid];          // channel 2

        f32x4 o0, o1, o2;
        #pragma unroll
        for (int k = 0; k < 4; ++k) {
            float r0, r1, r2;
            quant3(a0[k], a1[k], a2[k], r0, r1, r2);
            o0[k] = r0;
            o1[k] = r1;
            o2[k] = r2;
        }

        int b   = t >> 8;
        int col = (t & 255) * CHUNK;
        float* op = out + (size_t)b * 3 * HW + col;
        // write-once output: non-temporal so the input stays hot in the 192MB L2
        __builtin_nontemporal_store(o0, (f32x4*)(op) + tid);
        __builtin_nontemporal_store(o1, (f32x4*)(op + HW) + tid);
        __builtin_nontemporal_store(o2, (f32x4*)(op + 2 * HW) + tid);

        __syncthreads();                               // all reads of buf[cur] done before
        cur ^= 1;                                      // next iter's DMA overwrites it
    }
}

extern "C" void kernel_launch(void* const* d_in, const int* in_sizes, int n_in,
                              void* d_out, int out_size, void* d_ws, size_t ws_size,
                              hipStream_t stream) {
    (void)in_sizes; (void)n_in; (void)out_size; (void)d_ws; (void)ws_size;
    const float* x = (const float*)d_in[0];
    float*       y = (float*)d_out;
    ColorQuantization_kernel<<<dim3(GRID), dim3(THREADS), 0, stream>>>(x, y);
}